// JointAtt_17626545783396
// MI455X (gfx1250) — compile-verified
//
#include <hip/hip_runtime.h>
#include <hip/hip_bf16.h>

typedef __attribute__((ext_vector_type(2))) float v2f;
typedef __attribute__((ext_vector_type(4))) float v4f;
typedef __attribute__((ext_vector_type(8))) float v8f;

#define EPS 1e-5f

// ---------------------------------------------------------------------------
// Kernel 1: per-image row means (gh) and column means (gw).
// Grid: 8192 blocks = (n*512 + g*128 + c); block = 256 threads.
// Writes y[b][0..63] = row means (over w), y[b][64..127] = col means (over h).
// Note (n*4+g)*128 + c == n*512 + g*128 + c == blockIdx.x, so y rows index by b.
// ---------------------------------------------------------------------------
__global__ __launch_bounds__(256) void rowcol_mean_kernel(
    const float* __restrict__ x, float* __restrict__ y0) {
  __shared__ float img[64][65];  // pad to 65 -> conflict-free row & col sweeps
  const int b = blockIdx.x;
  const float* __restrict__ X = x + (size_t)b * 4096;
  const int t = threadIdx.x;

  // Coalesced load of the 16 KB image into LDS (1024 float4s).
  for (int i = t; i < 1024; i += 256) {
    v4f v = ((const v4f*)X)[i];
    int r = i >> 4, cq = (i & 15) << 2;
    img[r][cq + 0] = v.x; img[r][cq + 1] = v.y;
    img[r][cq + 2] = v.z; img[r][cq + 3] = v.w;
  }
  __syncthreads();

  if (t < 128) {
    float s = 0.0f;
    if (t < 64) {                       // row mean for h = t  (mean over w)
      for (int j = 0; j < 64; ++j) s += img[t][j];
      y0[(size_t)b * 128 + t] = s * (1.0f / 64.0f);
    } else {                            // col mean for w = t-64 (mean over h)
      int cc = t - 64;
      for (int r = 0; r < 64; ++r) s += img[r][cc];
      y0[(size_t)b * 128 + 64 + cc] = s * (1.0f / 64.0f);
    }
  }
}

// ---------------------------------------------------------------------------
// Kernel 2: fused MLP chain per (n,g), using V_WMMA_F32_16X16X4_F32.
// Grid: 64 blocks; block = 256 threads (8 waves).
// Phase A: wave w computes 16x16 l-tile w of  y1 = W1[16x128] * Y[128x128],
//          fuses bias + BN + hardswish, stages into LDS.
// Phase B: wave w owns c-rows [16w,16w+16) and sweeps all 8 l-tiles of
//          a = sigmoid(W{h,w}[128x16] * y1[16x64] + b), writing a_h/a_w.
// WMMA f32 16x16x4 layouts (ISA 7.12.2):
//   A: lane L, vgpr v -> A[L%16][k0 + 2*(L/16) + v]
//   B: lane L, vgpr v -> B[k0 + 2*(L/16) + v][L%16]
//   D: lane L, vgpr v -> D[v + 8*(L/16)][L%16]
// ---------------------------------------------------------------------------
__global__ __launch_bounds__(256) void mlp_att_kernel(
    const float* __restrict__ y0,    // [64][128(k=c)][128(l)]
    const float* __restrict__ W1,    // [16][128]
    const float* __restrict__ b1,    // [16]
    const float* __restrict__ gamma, const float* __restrict__ beta,
    const float* __restrict__ mean,  const float* __restrict__ var,
    const float* __restrict__ Wh,    // [128][16]
    const float* __restrict__ bh,    // [128]
    const float* __restrict__ Ww,    // [128][16]
    const float* __restrict__ bw,    // [128]
    float* __restrict__ a_h,         // [64][128][64]
    float* __restrict__ a_w) {       // [64][128][64]
  __shared__ float yl[16][130];      // y1 tile staged for phase B (pitch 130)

  const int ng   = blockIdx.x;       // 0..63
  const int wave = threadIdx.x >> 5; // 0..7
  const int lane = threadIdx.x & 31;
  const int lh   = lane >> 4;        // 0/1 : lane half
  const int lm   = lane & 15;        // 0..15

  const float* __restrict__ Y = y0 + (size_t)ng * 128 * 128;

  // ---- Phase A ----
  {
    const int l0 = wave * 16;
    v8f acc = {};
    for (int k0 = 0; k0 < 128; k0 += 4) {
      const int ka = k0 + 2 * lh;
      v2f a, bm;
      a.x  = W1[lm * 128 + ka];
      a.y  = W1[lm * 128 + ka + 1];
      bm.x = Y[(size_t)ka * 128 + l0 + lm];
      bm.y = Y[(size_t)(ka + 1) * 128 + l0 + lm];
      acc = __builtin_amdgcn_wmma_f32_16x16x4_f32(
          false, a, false, bm, (short)0, acc, false, false);
    }
    // epilogue: +b1, BN affine, hardswish; stage to LDS
    #pragma unroll
    for (int v = 0; v < 8; ++v) {
      const int m = v + 8 * lh;
      const float sc = gamma[m] * rsqrtf(var[m] + EPS);
      const float sh = beta[m] - mean[m] * sc;
      float yv = (acc[v] + b1[m]) * sc + sh;
      float t  = fminf(fmaxf(yv + 3.0f, 0.0f), 6.0f);
      yv = yv * t * (1.0f / 6.0f);
      yl[m][l0 + lm] = yv;
    }
  }
  __syncthreads();

  // ---- Phase B ----
  {
    const int c0 = wave * 16;
    for (int lt = 0; lt < 8; ++lt) {       // uniform across block: EXEC stays full
      const float* __restrict__ Wm = (lt < 4) ? Wh : Ww;
      const float* __restrict__ bm = (lt < 4) ? bh : bw;
      float* __restrict__ Am       = (lt < 4) ? a_h : a_w;
      const int l0 = lt * 16;
      const int lo = (lt < 4) ? l0 : (l0 - 64);
      v8f acc = {};
      #pragma unroll
      for (int k0 = 0; k0 < 16; k0 += 4) {
        const int ka = k0 + 2 * lh;
        v2f a, bb;
        a.x  = Wm[(c0 + lm) * 16 + ka];
        a.y  = Wm[(c0 + lm) * 16 + ka + 1];
        bb.x = yl[ka][l0 + lm];
        bb.y = yl[ka + 1][l0 + lm];
        acc = __builtin_amdgcn_wmma_f32_16x16x4_f32(
            false, a, false, bb, (short)0, acc, false, false);
      }
      #pragma unroll
      for (int v = 0; v < 8; ++v) {
        const int c = c0 + v + 8 * lh;
        float z = acc[v] + bm[c];
        z = 1.0f / (1.0f + __expf(-z));
        Am[((size_t)ng * 128 + c) * 64 + lo + lm] = z;
      }
    }
  }
}

// ---------------------------------------------------------------------------
// Kernel 3: out[n][co][h][w] = x[n][g*128+c][h][w] * a_h[b][h] * a_w[b][w],
// co = (c%4)*128 + g*32 + c/4 (channel shuffle folded into output pointer).
// Non-temporal load of x (last use) and non-temporal store of out so the
// output stream does not evict x from L2 (x fits: 128 MiB < 192 MB L2).
// ---------------------------------------------------------------------------
__global__ __launch_bounds__(256) void apply_att_kernel(
    const float* __restrict__ x,
    const float* __restrict__ a_h,   // [8192][64] indexed by b
    const float* __restrict__ a_w,   // [8192][64] indexed by b
    float* __restrict__ out) {
  __shared__ float sah[64];
  __shared__ v4f   saw[16];

  const int b = blockIdx.x;               // n*512 + g*128 + c
  const int c = b & 127, g = (b >> 7) & 3, n = b >> 9;
  const int co = (c & 3) * 128 + g * 32 + (c >> 2);

  const v4f* __restrict__ X = (const v4f*)(x + (size_t)b * 4096);
  v4f* __restrict__ O = (v4f*)(out + ((size_t)n * 512 + co) * 4096);

  const int t = threadIdx.x;
  if (t < 64) sah[t] = a_h[(size_t)b * 64 + t];
  else if (t < 80) saw[t - 64] = ((const v4f*)(a_w + (size_t)b * 64))[t - 64];
  __syncthreads();

  for (int i = t; i < 1024; i += 256) {
    const int r = i >> 4, q = i & 15;
    v4f v  = __builtin_nontemporal_load(&X[i]);
    const float ah = sah[r];
    v4f aw = saw[q];
    v4f o;
    o.x = v.x * ah * aw.x;
    o.y = v.y * ah * aw.y;
    o.z = v.z * ah * aw.z;
    o.w = v.w * ah * aw.w;
    __builtin_nontemporal_store(o, &O[i]);
  }
}

// ---------------------------------------------------------------------------
extern "C" void kernel_launch(void* const* d_in, const int* in_sizes, int n_in,
                              void* d_out, int out_size, void* d_ws, size_t ws_size,
                              hipStream_t stream) {
  const float* x     = (const float*)d_in[0];
  const float* W1    = (const float*)d_in[1];
  const float* b1    = (const float*)d_in[2];
  const float* gamma = (const float*)d_in[3];
  const float* beta  = (const float*)d_in[4];
  const float* mean  = (const float*)d_in[5];
  const float* var   = (const float*)d_in[6];
  const float* Wh    = (const float*)d_in[7];
  const float* bh    = (const float*)d_in[8];
  const float* Ww    = (const float*)d_in[9];
  const float* bw    = (const float*)d_in[10];

  float* y0 = (float*)d_ws;                 // 8192 * 128 floats = 4 MiB
  float* ah = y0 + (size_t)8192 * 128;      // 64*128*64 floats  = 2 MiB
  float* aw = ah + (size_t)64 * 128 * 64;   // 64*128*64 floats  = 2 MiB

  rowcol_mean_kernel<<<8192, 256, 0, stream>>>(x, y0);
  mlp_att_kernel<<<64, 256, 0, stream>>>(y0, W1, b1, gamma, beta, mean, var,
                                         Wh, bh, Ww, bw, ah, aw);
  apply_att_kernel<<<8192, 256, 0, stream>>>(x, ah, aw, (float*)d_out);
}